// DistanceInvLoss_15522011807780
// MI455X (gfx1250) — compile-verified
//
#include <hip/hip_runtime.h>
#include <math.h>

// ---------------------------------------------------------------------------
// TM-proximity loss on gfx1250 (MI455X).
// Pairwise sq-distances via Gram trick: ||xi-xj||^2 = ni + nj - 2*xi.xj,
// with the 16x16 tile of dot products from V_WMMA_F32_16X16X4_F32
// (A = 16x4 row coords, K=3 padded to 0; B = 4x16 col coords).
// Each wave owns one row-tile and sweeps TJW consecutive col-tiles, reusing
// its A fragments and row-side norms/masks.  Epilogue uses hardware
// v_sqrt_f32 / v_rcp_f32 (the VALU transcendental path is the roofline here:
// inputs are 48 KB, fully cache-resident, so HBM never matters).
// ---------------------------------------------------------------------------

typedef __attribute__((ext_vector_type(2))) float v2f;
typedef __attribute__((ext_vector_type(8))) float v8f;

#define NB    2          // batches
#define NPTS  2048       // 512 residues * 4 atoms
#define NT    (NPTS/16)  // 128 tiles per matrix dimension
#define TJW   4          // col-tiles handled per wave
#define NTJ   (NT/TJW)   // 32 wave-columns

// --- precompute squared norms of every point (both tensors) -----------------
__global__ void tm_norms_kernel(const float* __restrict__ pred,
                                const float* __restrict__ nat,
                                float* __restrict__ normP,
                                float* __restrict__ normN) {
  int t = blockIdx.x * blockDim.x + threadIdx.x;   // 0 .. NB*NPTS-1
  if (t >= NB * NPTS) return;
  const float* p = pred + 3 * t;
  const float* q = nat  + 3 * t;
  normP[t] = p[0]*p[0] + p[1]*p[1] + p[2]*p[2];
  normN[t] = q[0]*q[0] + q[1]*q[1] + q[2]*q[2];
}

// --- main: one wave32 per (row-tile, 4 col-tiles) strip ---------------------
__global__ void tm_tile_kernel(const float* __restrict__ pred,
                               const float* __restrict__ nat,
                               const unsigned char* __restrict__ mask, // bool_
                               const float* __restrict__ normP,
                               const float* __restrict__ normN,
                               double* __restrict__ accum,   // [0]=sum prox, [1]=count
                               float inv_d0sq) {
  const int gtid = blockIdx.x * blockDim.x + threadIdx.x;
  const int wave = gtid >> 5;
  const int lane = threadIdx.x & 31;

  const int b   = wave / (NT * NTJ);
  const int rem = wave % (NT * NTJ);
  const int ti  = rem / NTJ;              // row tile
  const int tj0 = (rem % NTJ) * TJW;      // first col tile of this strip

  const int m    = lane & 15;             // M (A) / N (B) index within tile
  const int half = lane >> 4;             // 0 -> K=0,1 ; 1 -> K=2,3
  const int base = b * NPTS;
  const int row  = ti * 16 + m;
  const int k0   = half * 2;
  const float hsel = (half == 0) ? 1.0f : 0.0f;   // kill K=3 pad lane

  // A fragments: unconditional loads (always in-bounds), select via multiply
  // so EXEC stays all-ones everywhere (WMMA requirement, and no exec churn).
  const float* pr = pred + 3 * (base + row);
  const float* nr = nat  + 3 * (base + row);
  v2f aP, aN;
  aP.x = pr[k0];  aP.y = pr[1] * hsel;
  aN.x = nr[k0];  aN.y = nr[1] * hsel;

  // Row-side epilogue constants: C/D element v in this lane has M = v + 8*half
  float nPi[8], nNi[8], mi[8];
#pragma unroll
  for (int v = 0; v < 8; ++v) {
    const int rowv = ti * 16 + v + 8 * half;
    nPi[v] = normP[base + rowv];          // uniform across 16 lanes -> broadcast
    nNi[v] = normN[base + rowv];
    mi[v]  = (float)mask[base + rowv];
  }

  float sumProx = 0.0f;
  float cnt     = 0.0f;

  for (int jt = 0; jt < TJW; ++jt) {
    const int col = (tj0 + jt) * 16 + m;
    const float* pc = pred + 3 * (base + col);
    const float* nc = nat  + 3 * (base + col);
    v2f bP, bN;
    bP.x = pc[k0];  bP.y = pc[1] * hsel;
    bN.x = nc[k0];  bN.y = nc[1] * hsel;

    v8f zero = {};
    v8f gP = __builtin_amdgcn_wmma_f32_16x16x4_f32(
        false, aP, false, bP, (short)0, zero, false, false);
    v8f gN = __builtin_amdgcn_wmma_f32_16x16x4_f32(
        false, aN, false, bN, (short)0, zero, false, false);

    const float nPj = normP[base + col];
    const float nNj = normN[base + col];
    const float mj  = (float)mask[base + col];

#pragma unroll
    for (int v = 0; v < 8; ++v) {
      const float dp2 = fmaxf(nPi[v] + nPj - 2.0f * gP[v], 0.0f);
      const float dn2 = fmaxf(nNi[v] + nNj - 2.0f * gN[v], 0.0f);
      const float dlt = __builtin_amdgcn_sqrtf(dp2)
                      - __builtin_amdgcn_sqrtf(dn2);          // v_sqrt_f32
      const float prox = __builtin_amdgcn_rcpf(
          fmaf(dlt * dlt, inv_d0sq, 1.0f));                   // v_rcp_f32
      const float w = mi[v] * mj;
      sumProx = fmaf(w, prox, sumProx);
      cnt    += w;
    }
  }

  // wave32 butterfly reduction
#pragma unroll
  for (int off = 16; off > 0; off >>= 1) {
    sumProx += __shfl_xor(sumProx, off, 32);
    cnt     += __shfl_xor(cnt,     off, 32);
  }

  if (lane == 0) {
    atomicAdd(&accum[0], (double)sumProx);   // global_atomic_add_f64
    atomicAdd(&accum[1], (double)cnt);
  }
}

// --- finalize: scalar result ------------------------------------------------
__global__ void tm_finalize_kernel(const double* __restrict__ accum,
                                   float* __restrict__ out) {
  out[0] = (float)(-accum[0] / accum[1]);
}

extern "C" void kernel_launch(void* const* d_in, const int* in_sizes, int n_in,
                              void* d_out, int out_size, void* d_ws, size_t ws_size,
                              hipStream_t stream) {
  const float* pred = (const float*)d_in[0];          // [2,512,4,3] f32
  const float* nat  = (const float*)d_in[1];          // [2,512,4,3] f32
  const unsigned char* mask = (const unsigned char*)d_in[2]; // [2,512,4] bool
  float* out = (float*)d_out;

  // workspace layout: [0..63] accumulators (2 doubles), then norms
  double* accum = (double*)d_ws;
  float* normP  = (float*)((char*)d_ws + 64);
  float* normN  = normP + NB * NPTS;

  hipMemsetAsync(d_ws, 0, 64, stream);   // zero accumulators (graph-capturable)

  // d0 = 1.24*(n_res-15)^(1/3) - 1.8 ; pass 1/d0^2
  const float d0 = 1.24f * cbrtf(512.0f - 15.0f) - 1.8f;
  const float inv_d0sq = 1.0f / (d0 * d0);

  // norms: NB*NPTS = 4096 points
  tm_norms_kernel<<<(NB * NPTS + 255) / 256, 256, 0, stream>>>(
      pred, nat, normP, normN);

  // waves: NB * NT * NTJ = 8192 ; 8 waves (256 threads) per block -> 1024 blocks
  const int n_waves  = NB * NT * NTJ;
  const int n_blocks = (n_waves * 32) / 256;
  tm_tile_kernel<<<n_blocks, 256, 0, stream>>>(
      pred, nat, mask, normP, normN, accum, inv_d0sq);

  tm_finalize_kernel<<<1, 1, 0, stream>>>(accum, out);
}